// Fusion_89008902242319
// MI455X (gfx1250) — compile-verified
//
#include <hip/hip_runtime.h>
#include <hip/hip_bf16.h>
#include <math.h>

#define D 128

typedef float v2f __attribute__((ext_vector_type(2)));
typedef float v8f __attribute__((ext_vector_type(8)));

// ---------------------------------------------------------------------------
// float atomic max via signed-max (val>=0) / unsigned-min (val<0) bit trick.
// Valid for IEEE-754; init value is -inf (0xFF800000).
// ---------------------------------------------------------------------------
__device__ __forceinline__ void atomicMaxF(float* addr, float val) {
    if (val >= 0.0f) atomicMax((int*)addr, __float_as_int(val));
    else             atomicMin((unsigned int*)addr, __float_as_uint(val));
}

__global__ void fill_f32(float* __restrict__ p, float v, int n) {
    int i = blockIdx.x * blockDim.x + threadIdx.x;
    if (i < n) p[i] = v;
}

// ---------------------------------------------------------------------------
// Z = H @ W using full-precision V_WMMA_F32_16X16X4_F32.
// H is a virtual concat: row r -> (r < n0) ? h0[r] : h1[r - n0].
// Block = 256 threads = 8 waves; wave w owns the 16-col tile w; block owns a
// 16-row strip. K-dim (128) walked in steps of 4 (32 chained WMMAs).
// ISA layouts (7.12.2, 32-bit):
//   A 16x4 : lanes 0-15 -> K=k0,k0+1 ; lanes 16-31 -> K=k0+2,k0+3 (M = lane%16)
//   B 4x16 : lanes 0-15 -> K=k0,k0+1 ; lanes 16-31 -> K=k0+2,k0+3 (N = lane%16)
//   C 16x16: VGPR v -> M = v + 8*(lane/16), N = lane%16
// ---------------------------------------------------------------------------
__global__ void gemm_z(const float* __restrict__ h0, int n0,
                       const float* __restrict__ h1,
                       const float* __restrict__ W,
                       float* __restrict__ Z, int n)
{
    const int wave = threadIdx.x >> 5;   // 0..7 -> column tile
    const int lane = threadIdx.x & 31;
    const int half = lane >> 4;          // 0: lanes 0-15, 1: lanes 16-31
    const int lr   = lane & 15;
    const int row0 = blockIdx.x * 16;
    const int col0 = wave * 16;

    int rowA = row0 + lr;
    if (rowA >= n) rowA = n - 1;         // clamp reads; store is guarded
    const float* hrow = (rowA < n0) ? (h0 + (size_t)rowA * D)
                                    : (h1 + (size_t)(rowA - n0) * D);

    v8f c = {0.f, 0.f, 0.f, 0.f, 0.f, 0.f, 0.f, 0.f};
#pragma unroll
    for (int k0 = 0; k0 < D; k0 += 4) {
        const int kk = k0 + 2 * half;
        float2 av = *(const float2*)(hrow + kk);
        v2f a; a.x = av.x; a.y = av.y;
        v2f b; b.x = W[(size_t)kk * D + col0 + lr];
               b.y = W[(size_t)(kk + 1) * D + col0 + lr];
        c = __builtin_amdgcn_wmma_f32_16x16x4_f32(
                false, a, false, b, (short)0, c, false, false);
    }
#pragma unroll
    for (int v = 0; v < 8; ++v) {
        const int r = row0 + v + 8 * half;
        if (r < n) Z[(size_t)r * D + col0 + lr] = c[v];
    }
}

// ---------------------------------------------------------------------------
// Per-row attention projections: s_src[i] = z[i].a[:D], s_dst[i] = z[i].a[D:]
// One wave32 per row; float4 loads; shuffle reduction.
// ---------------------------------------------------------------------------
__global__ void row_scores(const float* __restrict__ Z, const float* __restrict__ a,
                           float* __restrict__ ssrc, float* __restrict__ sdst, int n)
{
    const int wave = (blockIdx.x * blockDim.x + threadIdx.x) >> 5;
    const int lane = threadIdx.x & 31;
    if (wave >= n) return;
    const float4 zv = *(const float4*)(Z + (size_t)wave * D + lane * 4);
    const float4 a0 = *(const float4*)(a + lane * 4);
    const float4 a1 = *(const float4*)(a + D + lane * 4);
    float p0 = zv.x * a0.x + zv.y * a0.y + zv.z * a0.z + zv.w * a0.w;
    float p1 = zv.x * a1.x + zv.y * a1.y + zv.z * a1.z + zv.w * a1.w;
#pragma unroll
    for (int off = 16; off > 0; off >>= 1) {
        p0 += __shfl_down(p0, off, 32);
        p1 += __shfl_down(p1, off, 32);
    }
    if (lane == 0) { ssrc[wave] = p0; sdst[wave] = p1; }
}

__device__ __forceinline__ float leaky(float v) {
    return v > 0.0f ? v : 0.01f * v;
}

// Pass 1: segment max of leaky_relu edge logits over dst.
__global__ void edge_max(const int* __restrict__ src, const int* __restrict__ dst,
                         const float* __restrict__ ssrc, const float* __restrict__ sdst,
                         float* __restrict__ m, int nE)
{
    int e = blockIdx.x * blockDim.x + threadIdx.x;
    if (e >= nE) return;
    int d = dst[e];
    float v = leaky(ssrc[src[e]] + sdst[d]);
    atomicMaxF(&m[d], v);
}

// Pass 2: den[dst] += exp(e - m[dst])
__global__ void edge_den(const int* __restrict__ src, const int* __restrict__ dst,
                         const float* __restrict__ ssrc, const float* __restrict__ sdst,
                         const float* __restrict__ m, float* __restrict__ den, int nE)
{
    int e = blockIdx.x * blockDim.x + threadIdx.x;
    if (e >= nE) return;
    int d = dst[e];
    float v = leaky(ssrc[src[e]] + sdst[d]);
    atomicAdd(&den[d], __expf(v - m[d]));
}

// Pass 3: out[dst] += alpha * z[src]. One wave32 per edge; each lane owns a
// float4 slice (float4 gather + 4 global_atomic_add_f32).
// Output row r goes to out_lo[r] if r < n0 else out_hi[r - n0].
__global__ void edge_agg(const int* __restrict__ src, const int* __restrict__ dst,
                         const float* __restrict__ ssrc, const float* __restrict__ sdst,
                         const float* __restrict__ m, const float* __restrict__ den,
                         const float* __restrict__ z,
                         float* __restrict__ out_lo, float* __restrict__ out_hi,
                         int n0, int nE)
{
    int t = blockIdx.x * blockDim.x + threadIdx.x;
    int e = t >> 5;
    int lane = t & 31;
    if (e >= nE) return;
    int s = src[e], d = dst[e];
    float v = leaky(ssrc[s] + sdst[d]);
    float alpha = __expf(v - m[d]) / den[d];
    float4 zv = *(const float4*)(z + (size_t)s * D + lane * 4);
    float* op = ((d < n0) ? (out_lo + (size_t)d * D)
                          : (out_hi + (size_t)(d - n0) * D)) + lane * 4;
    atomicAdd(op + 0, alpha * zv.x);
    atomicAdd(op + 1, alpha * zv.y);
    atomicAdd(op + 2, alpha * zv.z);
    atomicAdd(op + 3, alpha * zv.w);
}

// ---------------------------------------------------------------------------
// Final fusion: per knowledge node i (one 128-thread block):
//   s1 = sum_j tanh((A_i @ W1)_j + b1_j) * w2_j ; s2 likewise with B_i
//   alpha = softmax([s1,s2])[0] ; out_i = alpha*A_i + (1-alpha)*B_i
// W1 column reads are coalesced across threads (thread j reads W1[k*D+j]).
// ---------------------------------------------------------------------------
__global__ void combine(const float* __restrict__ A, const float* __restrict__ Bfull,
                        const float* __restrict__ W1, const float* __restrict__ b1,
                        const float* __restrict__ w2, float* __restrict__ out,
                        int Isplit)
{
    __shared__ float sA[D], sB[D], red[D];
    const int i = blockIdx.x;
    const int j = threadIdx.x;
    sA[j] = A[(size_t)i * D + j];
    sB[j] = Bfull[(size_t)(Isplit + i) * D + j];
    __syncthreads();
    float accA = b1[j], accB = accA;
    for (int k = 0; k < D; ++k) {
        float wv = W1[(size_t)k * D + j];
        accA = fmaf(sA[k], wv, accA);
        accB = fmaf(sB[k], wv, accB);
    }
    const float w2j = w2[j];
    const float tA = tanhf(accA) * w2j;
    const float tB = tanhf(accB) * w2j;

    red[j] = tA; __syncthreads();
    for (int s = 64; s > 0; s >>= 1) { if (j < s) red[j] += red[j + s]; __syncthreads(); }
    const float s1 = red[0]; __syncthreads();
    red[j] = tB; __syncthreads();
    for (int s = 64; s > 0; s >>= 1) { if (j < s) red[j] += red[j + s]; __syncthreads(); }
    const float s2 = red[0];

    const float mm = fmaxf(s1, s2);
    const float e1 = __expf(s1 - mm), e2 = __expf(s2 - mm);
    const float al = e1 / (e1 + e2);
    out[(size_t)i * D + j] = al * sA[j] + (1.0f - al) * sB[j];
}

// ---------------------------------------------------------------------------
static inline int cdiv(int a, int b) { return (a + b - 1) / b; }

extern "C" void kernel_launch(void* const* d_in, const int* in_sizes, int n_in,
                              void* d_out, int out_size, void* d_ws, size_t ws_size,
                              hipStream_t stream)
{
    (void)n_in; (void)ws_size;
    const float* kn_emb   = (const float*)d_in[0];
    const float* exer_emb = (const float*)d_in[1];
    const float* W_dir    = (const float*)d_in[2];
    const float* a_dir    = (const float*)d_in[3];
    const float* W_ke     = (const float*)d_in[4];
    const float* a_ke     = (const float*)d_in[5];
    const float* W_ek     = (const float*)d_in[6];
    const float* a_ek     = (const float*)d_in[7];
    const float* rW1      = (const float*)d_in[8];
    const float* rb1      = (const float*)d_in[9];
    const float* rw2      = (const float*)d_in[10];
    const int* dsrc  = (const int*)d_in[11];
    const int* ddst  = (const int*)d_in[12];
    const int* kesrc = (const int*)d_in[13];
    const int* kedst = (const int*)d_in[14];
    const int* eksrc = (const int*)d_in[15];
    const int* ekdst = (const int*)d_in[16];

    const int Kn = in_sizes[0] / D;
    const int In = in_sizes[1] / D;
    const int Nn = Kn + In;
    const int E1 = in_sizes[11];
    const int E2 = in_sizes[13];

    // workspace carve-out (~83 MB in fp32)
    float* w = (float*)d_ws;
    size_t off = 0;
    float* buf_z = w + off; off += (size_t)Nn * D;   // z for current GAT
    float* buf_B = w + off; off += (size_t)Nn * D;   // ke-GAT accumulator (rows I.. used)
    float* buf_A = w + off; off += (size_t)Kn * D;   // dir-GAT acc / ek-GAT dump
    float* ssrc  = w + off; off += (size_t)Nn;
    float* sdst  = w + off; off += (size_t)Nn;
    float* mbuf  = w + off; off += (size_t)Nn;
    float* dbuf  = w + off; off += (size_t)Nn;

    float* out_kn = (float*)d_out;                       // [K, D]
    float* out_ex = (float*)d_out + (size_t)Kn * D;      // [I, D]

    hipMemsetAsync(d_out, 0, (size_t)out_size * sizeof(float), stream);

    // ================= GAT 3 (ek): Cfull; rows < I straight into d_out ======
    gemm_z<<<cdiv(Nn, 16), 256, 0, stream>>>(exer_emb, In, kn_emb, W_ek, buf_z, Nn);
    row_scores<<<cdiv(Nn, 8), 256, 0, stream>>>(buf_z, a_ek, ssrc, sdst, Nn);
    fill_f32<<<cdiv(Nn, 256), 256, 0, stream>>>(mbuf, -INFINITY, Nn);
    hipMemsetAsync(dbuf, 0, (size_t)Nn * sizeof(float), stream);
    hipMemsetAsync(buf_A, 0, (size_t)Kn * D * sizeof(float), stream);  // dump rows >= I
    edge_max<<<cdiv(E2, 256), 256, 0, stream>>>(eksrc, ekdst, ssrc, sdst, mbuf, E2);
    edge_den<<<cdiv(E2, 256), 256, 0, stream>>>(eksrc, ekdst, ssrc, sdst, mbuf, dbuf, E2);
    edge_agg<<<cdiv(E2 * 32, 256), 256, 0, stream>>>(eksrc, ekdst, ssrc, sdst, mbuf, dbuf,
                                                     buf_z, out_ex, buf_A, In, E2);

    // ================= GAT 2 (ke): Bfull into buf_B =========================
    gemm_z<<<cdiv(Nn, 16), 256, 0, stream>>>(exer_emb, In, kn_emb, W_ke, buf_z, Nn);
    row_scores<<<cdiv(Nn, 8), 256, 0, stream>>>(buf_z, a_ke, ssrc, sdst, Nn);
    fill_f32<<<cdiv(Nn, 256), 256, 0, stream>>>(mbuf, -INFINITY, Nn);
    hipMemsetAsync(dbuf, 0, (size_t)Nn * sizeof(float), stream);
    hipMemsetAsync(buf_B, 0, (size_t)Nn * D * sizeof(float), stream);
    edge_max<<<cdiv(E2, 256), 256, 0, stream>>>(kesrc, kedst, ssrc, sdst, mbuf, E2);
    edge_den<<<cdiv(E2, 256), 256, 0, stream>>>(kesrc, kedst, ssrc, sdst, mbuf, dbuf, E2);
    edge_agg<<<cdiv(E2 * 32, 256), 256, 0, stream>>>(kesrc, kedst, ssrc, sdst, mbuf, dbuf,
                                                     buf_z, buf_B, buf_B, Nn, E2);

    // ================= GAT 1 (directed, K nodes): A into buf_A ==============
    gemm_z<<<cdiv(Kn, 16), 256, 0, stream>>>(kn_emb, Kn, kn_emb, W_dir, buf_z, Kn);
    row_scores<<<cdiv(Kn, 8), 256, 0, stream>>>(buf_z, a_dir, ssrc, sdst, Kn);
    fill_f32<<<cdiv(Kn, 256), 256, 0, stream>>>(mbuf, -INFINITY, Kn);
    hipMemsetAsync(dbuf, 0, (size_t)Kn * sizeof(float), stream);
    hipMemsetAsync(buf_A, 0, (size_t)Kn * D * sizeof(float), stream);
    edge_max<<<cdiv(E1, 256), 256, 0, stream>>>(dsrc, ddst, ssrc, sdst, mbuf, E1);
    edge_den<<<cdiv(E1, 256), 256, 0, stream>>>(dsrc, ddst, ssrc, sdst, mbuf, dbuf, E1);
    edge_agg<<<cdiv(E1 * 32, 256), 256, 0, stream>>>(dsrc, ddst, ssrc, sdst, mbuf, dbuf,
                                                     buf_z, buf_A, buf_A, Kn, E1);

    // ================= gated fusion -> kn_out ===============================
    combine<<<Kn, D, 0, stream>>>(buf_A, buf_B, rW1, rb1, rw2, out_kn, In);
}